// SE_loss_37099927503097
// MI455X (gfx1250) — compile-verified
//
#include <hip/hip_runtime.h>
#include <hip/hip_bf16.h>
#include <math.h>

typedef __attribute__((ext_vector_type(2))) float v2f;
typedef __attribute__((ext_vector_type(8))) float v8f;

#define NCLASS 6
#define NBATCH 64
#define SCAN_THREADS 256
#define VEC_PER_THREAD 16   // int4 (4 labels) per thread per loop iter x 16 iters

// ---------------------------------------------------------------------------
// Zero the per-batch presence masks in workspace (must happen every call:
// harness does not re-poison, and we accumulate with atomicOr).
// ---------------------------------------------------------------------------
__global__ void init_masks(unsigned* __restrict__ masks) {
    masks[threadIdx.x] = 0u;
}

// ---------------------------------------------------------------------------
// Streaming presence scan: memory-bound. Each thread loads
// int4 (16B, global_load_b128), ORs (1<<label) into a private mask, block
// reduces via LDS atomicOr (ds path), one global_atomic_or per block.
// blockIdx.y = batch index.
// ---------------------------------------------------------------------------
__global__ __launch_bounds__(SCAN_THREADS)
void presence_scan(const int* __restrict__ target,
                   unsigned* __restrict__ masks,
                   int vec_per_batch) {
    const int batch = blockIdx.y;
    const int4* __restrict__ t4 =
        reinterpret_cast<const int4*>(target) + (size_t)batch * vec_per_batch;

    unsigned m = 0u;
    const int base = blockIdx.x * (SCAN_THREADS * VEC_PER_THREAD) + threadIdx.x;

#pragma unroll
    for (int i = 0; i < VEC_PER_THREAD; ++i) {
        const int idx = base + i * SCAN_THREADS;
        if (idx < vec_per_batch) {
            // Speculative prefetch one stride ahead (global_prefetch_b8);
            // translation failures past the end are silently dropped.
            __builtin_prefetch(&t4[idx + SCAN_THREADS], 0, 0);
            const int4 v = t4[idx];
            m |= (1u << (v.x & 31)) | (1u << (v.y & 31)) |
                 (1u << (v.z & 31)) | (1u << (v.w & 31));
        }
    }

    __shared__ unsigned lmask;
    if (threadIdx.x == 0) lmask = 0u;
    __syncthreads();
    if (m) atomicOr(&lmask, m);
    __syncthreads();
    if (threadIdx.x == 0 && lmask) atomicOr(&masks[batch], lmask);
}

// ---------------------------------------------------------------------------
// BCE epilogue over 384 elements. 6 full waves, 2 elements per lane.
// Final sum runs through v_wmma_f32_16x16x4_f32: with B = ones,
// D[m][n] = rowsum(A)[m]; column N=0 lives in lanes 0 (M=0..7) and 16
// (M=8..15), so (sum of 8 D regs) + shfl_xor(16) = exact f32 wave total.
// No divergence before the WMMA -> EXEC is all ones as required.
// ---------------------------------------------------------------------------
__global__ __launch_bounds__(192)
void bce_finalize(const float* __restrict__ se_pred,
                  const unsigned* __restrict__ masks,
                  float* __restrict__ out,
                  int nelem) {
    const int tid = threadIdx.x;

    v2f a;
#pragma unroll
    for (int j = 0; j < 2; ++j) {
        const int e = tid * 2 + j;          // 0..383, always in range
        const int b = e / NCLASS;
        const int c = e % NCLASS;
        const float p  = se_pred[e];
        const float t  = (float)((masks[b] >> c) & 1u);
        const float lp  = fmaxf(logf(p),      -100.0f);   // torch log clamp
        const float l1p = fmaxf(log1pf(-p),   -100.0f);
        a[j] = -(t * lp + (1.0f - t) * l1p);
    }

    v2f bones;
    bones[0] = 1.0f;
    bones[1] = 1.0f;
    v8f cz = {};
    // D = A(16x4) * ones(4x16) + 0   (f32 WMMA, exact)
    v8f d = __builtin_amdgcn_wmma_f32_16x16x4_f32(
        /*neg_a=*/false, a, /*neg_b=*/false, bones,
        /*c_mod=*/(short)0, cz, /*reuse_a=*/false, /*reuse_b=*/false);

    float s = d[0] + d[1] + d[2] + d[3] + d[4] + d[5] + d[6] + d[7];
    s += __shfl_xor(s, 16, 32);             // combine M=0..7 and M=8..15 halves

    __shared__ float partial[6];
    const int wave = tid >> 5;
    const int lane = tid & 31;
    if (lane == 0) partial[wave] = s;
    __syncthreads();
    if (tid == 0) {
        float tot = 0.0f;
#pragma unroll
        for (int w = 0; w < 6; ++w) tot += partial[w];
        out[0] = tot / (float)nelem;
    }
}

// ---------------------------------------------------------------------------
extern "C" void kernel_launch(void* const* d_in, const int* in_sizes, int n_in,
                              void* d_out, int out_size, void* d_ws, size_t ws_size,
                              hipStream_t stream) {
    const float* se_pred = (const float*)d_in[0];     // [64,6] f32
    const int*   target  = (const int*)d_in[1];       // [64,512,512] labels
    unsigned*    masks   = (unsigned*)d_ws;           // 64 x u32 presence masks

    const int n_pred        = in_sizes[0];                    // 384
    const long long n_tgt   = (long long)in_sizes[1];         // 16777216
    const int per_batch     = (int)(n_tgt / NBATCH);          // 262144
    const int vec_per_batch = per_batch / 4;                  // 65536 int4

    init_masks<<<dim3(1), dim3(NBATCH), 0, stream>>>(masks);

    const int chunk = SCAN_THREADS * VEC_PER_THREAD;          // 4096 int4/block
    const int blocks_x = (vec_per_batch + chunk - 1) / chunk; // 16
    presence_scan<<<dim3(blocks_x, NBATCH), dim3(SCAN_THREADS), 0, stream>>>(
        target, masks, vec_per_batch);

    bce_finalize<<<dim3(1), dim3(192), 0, stream>>>(
        se_pred, masks, (float*)d_out, n_pred);
}